// Transformer_28930899705943
// MI455X (gfx1250) — compile-verified
//
#include <hip/hip_runtime.h>

// Embedding gather: out[t, e] = W[e, tokens[t]]
//   tokens: (B*S,) int32   = 4096
//   W:      (EMBED, VOCAB) = (1024, 50257) fp32, row-major
//   out:    (B*S, EMBED) fp32
//
// Pure data movement, memory-bound: ~191 MB of W cachelines + 16 MB out
// => ~9 us floor at 23.3 TB/s. WMMA is intentionally NOT used: the one-hot
// matmul is 422 GFLOP of zeros and strictly slower than the gather floor.
//
// CDNA5 path: asynchronous copy engine. Gathered elements move
// global -> LDS (GLOBAL_LOAD_ASYNC_TO_LDS_B32, per-lane scattered addresses)
// then LDS -> global (GLOBAL_STORE_ASYNC_FROM_LDS_B128, coalesced), tracked
// on ASYNCcnt. Data never transits VGPRs; the wave only issues addresses.
// TH_LOAD_NT_RT: bypass WGP$ (zero near reuse for strided columns), keep W
// in the 192 MB L2 (cross-block reuse). TH_STORE_NT: write-once output must
// not evict W's L2 working set.
//
// NOTE (CDNA5 ISA): INST_OFFSET is added to BOTH the LDS
// and global addresses of an async op, so each of the 4 strided gather loads
// (global stride 201028 B, LDS stride 4 B) needs its own address operands.

#define VOCAB 50257
#define EMBED 1024

__global__ __launch_bounds__(256) void embed_gather_async_kernel(
    const int* __restrict__ tokens,
    const float* __restrict__ W,
    float* __restrict__ out,
    int n_tokens)
{
    __shared__ __align__(16) float buf[EMBED];   // 4 KB staging tile per block

    const int t = blockIdx.x;               // one block per token
    if (t >= n_tokens) return;

    const int tok = tokens[t];              // uniform -> scalar load path
    const int tid = threadIdx.x;            // 0..255, each owns 4 embed rows

    // Per-lane global gather addresses: column `tok`, rows e = 4*tid + k.
    const float* g0 = W + (size_t)(tid * 4) * (size_t)VOCAB + (size_t)tok;
    const float* g1 = g0 + (size_t)VOCAB;
    const float* g2 = g1 + (size_t)VOCAB;
    const float* g3 = g2 + (size_t)VOCAB;

    // Lane-private LDS staging slot: bytes [16*tid, 16*tid+16).
    // Low 32 bits of a generic LDS pointer are the wave-relative LDS offset.
    const uint32_t lds0 = (uint32_t)(uintptr_t)buf + (uint32_t)(tid * 16);

    // Async gather: global -> LDS, 4 scattered B32 per lane (ASYNCcnt += 4).
    asm volatile("global_load_async_to_lds_b32 %0, %1, off th:TH_LOAD_NT_RT"
                 :: "v"(lds0),      "v"(g0) : "memory");
    asm volatile("global_load_async_to_lds_b32 %0, %1, off th:TH_LOAD_NT_RT"
                 :: "v"(lds0 + 4),  "v"(g1) : "memory");
    asm volatile("global_load_async_to_lds_b32 %0, %1, off th:TH_LOAD_NT_RT"
                 :: "v"(lds0 + 8),  "v"(g2) : "memory");
    asm volatile("global_load_async_to_lds_b32 %0, %1, off th:TH_LOAD_NT_RT"
                 :: "v"(lds0 + 12), "v"(g3) : "memory");

    // Wave-local completion of this wave's async loads into LDS. The staged
    // data is lane-private within the wave, so no workgroup barrier needed.
    asm volatile("s_wait_asynccnt 0x0" ::: "memory");

    // Async coalesced store: LDS -> global, 16 B per lane => each wave32
    // writes 512 contiguous bytes of out. s_endpgm's implicit wait-idle
    // covers the outstanding async store.
    float* o = out + (size_t)t * EMBED + (size_t)(tid * 4);
    asm volatile("global_store_async_from_lds_b128 %0, %1, off th:TH_STORE_NT"
                 :: "v"(o), "v"(lds0) : "memory");
}

extern "C" void kernel_launch(void* const* d_in, const int* in_sizes, int n_in,
                              void* d_out, int out_size, void* d_ws, size_t ws_size,
                              hipStream_t stream)
{
    const int*   tokens = (const int*)d_in[0];   // (B*S,) int32
    const float* W      = (const float*)d_in[1]; // (EMBED, VOCAB) fp32
    float*       out    = (float*)d_out;         // (B*S, EMBED) fp32

    const int n_tokens = in_sizes[0];            // B*S = 4096

    // One block per token: 256 threads x 4 floats = 1024 = EMBED.
    embed_gather_async_kernel<<<dim3(n_tokens), dim3(256), 0, stream>>>(
        tokens, W, out, n_tokens);
}